// Sobel_15547781611849
// MI455X (gfx1250) — compile-verified
//
#include <hip/hip_runtime.h>
#include <stdint.h>

#define IMG   512
#define CH    32
#define TH    8
#define TW    16
#define IN_H  (TH + 4)   // 12 rows staged (halo 2)
#define IN_W  (TW + 4)   // 20 cols staged
#define PL_H  (TH + 2)   // 10 pooled rows (halo 1)
#define PL_W  (TW + 2)   // 18 pooled cols

// lane = channel (wave32 == 32 channels): all global traffic is 128B/pixel
// coalesced, all LDS traffic bank-conflict-free (pixel stride = 128B).
__global__ __launch_bounds__(256) void sobel_pool_kernel(
    const float* __restrict__ x, float* __restrict__ out)
{
  __shared__ __attribute__((aligned(16))) float s_in[IN_H * IN_W * CH]; // 30720 B
  __shared__ __attribute__((aligned(16))) float s_pl[PL_H * PL_W * CH]; // 23040 B

  const int tid  = threadIdx.x;
  const int lane = tid & 31;
  const int wid  = tid >> 5;

  const int w0 = blockIdx.x * TW;
  const int h0 = blockIdx.y * TH;
  const int b  = blockIdx.z;

  // ---------------- Phase 1: async DMA input tile (halo 2) into LDS -------
  // One pixel = 32ch * 4B = 128 B = 8 x b128 transfers.
  const int NXFER = IN_H * IN_W * 8;   // 1920 16-byte transfers
  for (int t = tid; t < NXFER; t += 256) {
    const int pix = t >> 3;
    const int sub = t & 7;
    const int pr  = pix / IN_W;
    const int pc  = pix - pr * IN_W;
    const int h   = h0 - 2 + pr;
    const int w   = w0 - 2 + pc;
    float* ldst = &s_in[pix * CH + sub * 4];
    if ((unsigned)h < IMG && (unsigned)w < IMG) {
      const float* gsrc = x + (((size_t)b * IMG + h) * IMG + w) * CH + sub * 4;
      uint32_t loff  = (uint32_t)(uintptr_t)ldst;   // low 32b of generic ptr = LDS byte addr
      uint64_t gaddr = (uint64_t)(uintptr_t)gsrc;
      asm volatile("global_load_async_to_lds_b128 %0, %1, off"
                   :: "v"(loff), "v"(gaddr) : "memory");
    } else {
      *(float4*)ldst = make_float4(0.f, 0.f, 0.f, 0.f);  // zero padding
    }
  }
#if __has_builtin(__builtin_amdgcn_s_wait_asynccnt)
  __builtin_amdgcn_s_wait_asynccnt(0);
#else
  asm volatile("s_wait_asynccnt 0" ::: "memory");
#endif
  __syncthreads();

  // ---------------- Phase 2: 3x3 valid-count average pool (halo 1) --------
  for (int p = wid; p < PL_H * PL_W; p += 8) {
    const int pr = p / PL_W;
    const int pc = p - pr * PL_W;
    const int h  = h0 - 1 + pr;
    const int w  = w0 - 1 + pc;
    float v = 0.f;
    if ((unsigned)h < IMG && (unsigned)w < IMG) {
      float s = 0.f;
#pragma unroll
      for (int i = 0; i < 3; ++i)
#pragma unroll
        for (int j = 0; j < 3; ++j)
          s += s_in[((pr + i) * IN_W + (pc + j)) * CH + lane];
      const int rh = min(h + 1, IMG - 1) - max(h - 1, 0) + 1;  // 2 or 3
      const int cw = min(w + 1, IMG - 1) - max(w - 1, 0) + 1;  // 2 or 3
      v = s / (float)(rh * cw);
    }
    s_pl[p * CH + lane] = v;   // out-of-image pooled halo stays 0 (Sobel zero-pad)
  }
  __syncthreads();

  // ---------------- Phase 3: depthwise Sobel + magnitude ------------------
  for (int q = wid; q < TH * TW; q += 8) {
    const int qr = q >> 4;          // TW == 16
    const int qc = q & (TW - 1);
    const int base = (qr * PL_W + qc) * CH + lane;
    const float tl  = s_pl[base];
    const float tc_ = s_pl[base + CH];
    const float tr  = s_pl[base + 2 * CH];
    const float ml  = s_pl[base + PL_W * CH];
    const float mr  = s_pl[base + (PL_W + 2) * CH];
    const float bl  = s_pl[base + 2 * PL_W * CH];
    const float bc_ = s_pl[base + (2 * PL_W + 1) * CH];
    const float br  = s_pl[base + (2 * PL_W + 2) * CH];
    float gx = ((tl - tr) + 2.f * (ml - mr) + (bl - br)) * 0.25f;
    float gy = ((tl + 2.f * tc_ + tr) - (bl + 2.f * bc_ + br)) * 0.25f;
    const float r = sqrtf(fmaf(gx, gx, fmaf(gy, gy, 1e-6f)));
    out[(((size_t)b * IMG + (h0 + qr)) * IMG + (w0 + qc)) * CH + lane] = r;
  }
}

extern "C" void kernel_launch(void* const* d_in, const int* in_sizes, int n_in,
                              void* d_out, int out_size, void* d_ws, size_t ws_size,
                              hipStream_t stream) {
  (void)in_sizes; (void)n_in; (void)d_ws; (void)ws_size; (void)out_size;
  const float* x = (const float*)d_in[0];
  float* out = (float*)d_out;
  dim3 grid(IMG / TW, IMG / TH, 8);   // 32 x 64 x 8 = 16384 blocks
  dim3 block(256);                    // 8 waves
  hipLaunchKernelGGL(sobel_pool_kernel, grid, block, 0, stream, x, out);
}